// LongformerAttention_15590731284882
// MI455X (gfx1250) — compile-verified
//
#include <hip/hip_runtime.h>
#include <cmath>

// ---------------------------------------------------------------------------
// CDNA5 fp32 WMMA: V_WMMA_F32_16X16X4_F32
//   A: 16x4 f32  (2 VGPRs: lane<16 -> row=lane, K={0,1}; lane>=16 -> K={2,3})
//   B: 4x16 f32  (2 VGPRs: lane half selects K row pair, N = lane&15)
//   C/D: 16x16 f32 (8 VGPRs: vgpr r -> M=r (+8 for upper half), N = lane&15)
// ---------------------------------------------------------------------------
typedef float v2f __attribute__((ext_vector_type(2)));
typedef float v8f __attribute__((ext_vector_type(8)));

__device__ __forceinline__ v8f wmma_f32(v2f a, v2f b, v8f c) {
  // 8 args: (neg_a, A, neg_b, B, c_mod, C, reuse_a, reuse_b)
  return __builtin_amdgcn_wmma_f32_16x16x4_f32(false, a, false, b, (short)0, c,
                                               false, false);
}

#define SEQ  2048
#define DIM  512
#define NH   8
#define DH   64
#define MROWS 4096   // B*S

// ---------------------------------------------------------------------------
// C[M x 512] = A[M x 512] * W[512 x 512] + bias[512]
// block: 256 threads (8 waves), block tile 64(M) x 32(N), K staged 16 wide.
//  - A tile: one float4 per thread, register-prefetched one chunk ahead
//  - B tile: global_load_async_to_lds_b64 (CDNA5 async copy, ASYNCcnt)
// ---------------------------------------------------------------------------
__global__ __launch_bounds__(256) void gemm_512(const float* __restrict__ A,
                                                const float* __restrict__ W,
                                                const float* __restrict__ bias,
                                                float* __restrict__ C) {
  __shared__ __align__(16) float lA[64 * 16];
  __shared__ __align__(16) float lB[16 * 32];

  const int tid  = threadIdx.x;
  const int lane = tid & 31;
  const int wav  = tid >> 5;   // 0..7
  const int ms   = wav & 3;    // M sub-tile 0..3
  const int ns   = wav >> 2;   // N sub-tile 0..1
  const int l16  = lane & 15;
  const int half = lane >> 4;

  const int row0 = blockIdx.x * 64;
  const int col0 = blockIdx.y * 32;

  // A staging: thread -> (m = tid>>2, k4 = (tid&3)*4), one float4 each
  const int am = tid >> 2;
  const int ak = (tid & 3) << 2;
  const float* aptr = A + (size_t)(row0 + am) * DIM + ak;

  // B staging: thread -> (kk = tid>>4, n2 = (tid&15)*2), async b64 each
  const int bk = tid >> 4;
  const int bn = (tid & 15) << 1;
  const float* wptr = W + (size_t)bk * DIM + col0 + bn;
  const unsigned ldsB = (unsigned)(uintptr_t)(void*)&lB[bk * 32 + bn];

  float4 ra = *(const float4*)aptr;   // chunk 0 prefetch
  v8f acc = {};

  for (int k = 0; k < DIM; k += 16) {
    __syncthreads();                       // previous chunk fully consumed
    *(float4*)&lA[am * 16 + ak] = ra;      // ds_store_b128
    // direct global -> LDS async copy of the 16x32 B chunk
    asm volatile("global_load_async_to_lds_b64 %0, %1, off"
                 :: "v"(ldsB), "v"(wptr + (size_t)k * DIM)
                 : "memory");
    asm volatile("s_wait_asynccnt 0x0" ::: "memory");
    __syncthreads();                       // all waves staged

    if (k + 16 < DIM) ra = *(const float4*)(aptr + k + 16);  // overlap w/ MMA

#pragma unroll
    for (int t = 0; t < 4; ++t) {
      const int kk = t * 4 + half * 2;
      v2f a, b;
      a.x = lA[(ms * 16 + l16) * 16 + kk];
      a.y = lA[(ms * 16 + l16) * 16 + kk + 1];
      b.x = lB[kk * 32 + ns * 16 + l16];
      b.y = lB[(kk + 1) * 32 + ns * 16 + l16];
      acc = wmma_f32(a, b, acc);
    }
  }

  const int col = col0 + ns * 16 + l16;
  const float bv = bias[col];
#pragma unroll
  for (int r = 0; r < 8; ++r) {
    const int row = row0 + ms * 16 + r + half * 8;
    C[(size_t)row * DIM + col] = acc[r] + bv;
  }
}

// ---------------------------------------------------------------------------
// Flash-style masked attention, one wave per 16-query tile.
// grid: (S/16/4, B*H, 2 patterns). pattern 0 = sliding-window, 1 = global.
// Writes merged x[B,S,H,DH]: row stored only by the pattern matching gm[row].
// ---------------------------------------------------------------------------
__global__ __launch_bounds__(128) void attn_kernel(
    const float* __restrict__ q0, const float* __restrict__ k0,
    const float* __restrict__ v0, const float* __restrict__ q1,
    const float* __restrict__ k1, const float* __restrict__ v1,
    const int* __restrict__ gmask, float* __restrict__ X) {
  __shared__ float pbuf[4][16 * 17];   // stride 17: conflict-free A-frag reads

  const int lane = threadIdx.x & 31;
  const int w    = threadIdx.x >> 5;
  const int l16  = lane & 15;
  const int half = lane >> 4;

  const int qbase = (blockIdx.x * 4 + w) * 16;
  const int b  = blockIdx.y >> 3;
  const int hh = blockIdx.y & 7;
  const int pat = blockIdx.z;          // 0 = sw, 1 = g

  const float* Q = pat ? q1 : q0;
  const float* K = pat ? k1 : k0;
  const float* V = pat ? v1 : v0;

  const size_t bs = (size_t)b * SEQ;

  // preload Q fragments for all 16 K-steps (16x64 tile, 32 floats/lane)
  float aq[32];
  {
    const float* qrow = Q + (bs + qbase + l16) * DIM + hh * DH;
#pragma unroll
    for (int t = 0; t < 16; ++t) {
      const int kk = t * 4 + half * 2;
      aq[2 * t]     = qrow[kk];
      aq[2 * t + 1] = qrow[kk + 1];
    }
  }

  // global-mask bits for the 16 query rows of this tile
  const int gq_l = (gmask[bs + qbase + l16] != 0);
  const unsigned gqm = (unsigned)(__ballot(gq_l) & 0xFFFFu);

  float m_r[8], l_r[8];
  v8f acc[4];
#pragma unroll
  for (int r = 0; r < 8; ++r) { m_r[r] = -3.0e38f; l_r[r] = 0.f; }
#pragma unroll
  for (int nt = 0; nt < 4; ++nt) acc[nt] = {};

  for (int kt = 0; kt < SEQ / 16; ++kt) {
    const int key0 = kt * 16;
    const int gk = (gmask[bs + key0 + l16] != 0);

    // prefetch next key tile's K/V rows into cache (global_prefetch_b8)
    if (kt + 1 < SEQ / 16) {
      __builtin_prefetch(K + (bs + key0 + 16 + l16) * DIM + hh * DH, 0, 0);
      __builtin_prefetch(V + (bs + key0 + 16 + l16) * DIM + hh * DH, 0, 0);
    }

    // ---- S = Q * K^T (16x16), K-dim 64 via 16 chained WMMAs ----
    v8f s = {};
    const float* krow = K + (bs + key0 + l16) * DIM + hh * DH;
#pragma unroll
    for (int t = 0; t < 16; ++t) {
      const int kk = t * 4 + half * 2;
      v2f a, bf;
      a.x  = aq[2 * t];
      a.y  = aq[2 * t + 1];
      bf.x = krow[kk];
      bf.y = krow[kk + 1];
      s = wmma_f32(a, bf, s);
    }

    // ---- mask + online softmax (hardware v_exp via __expf) ----
    const int col = key0 + l16;
#pragma unroll
    for (int r = 0; r < 8; ++r) {
      const int rit = r + half * 8;
      const int row = qbase + rit;
      bool allowed = (((gqm >> rit) & 1u) != 0u) || (gk != 0);
      if (pat == 0) {
        const int d = row - col;
        allowed = allowed || (d >= -256 && d <= 255);
      }
      float sv = s[r] * 0.125f + (allowed ? 0.f : -1.0e10f);

      float mx = sv;
#pragma unroll
      for (int off = 8; off >= 1; off >>= 1)
        mx = fmaxf(mx, __shfl_xor(mx, off, 16));

      const float mn    = fmaxf(m_r[r], mx);
      const float alpha = __expf(m_r[r] - mn);
      const float p     = __expf(sv - mn);

      float ps = p;
#pragma unroll
      for (int off = 8; off >= 1; off >>= 1)
        ps += __shfl_xor(ps, off, 16);

      l_r[r] = l_r[r] * alpha + ps;
      m_r[r] = mn;
#pragma unroll
      for (int nt = 0; nt < 4; ++nt) acc[nt][r] *= alpha;

      pbuf[w][rit * 17 + l16] = p;   // stage P for A-fragment reload
    }
    asm volatile("s_wait_dscnt 0x0" ::: "memory");

    // ---- O += P * V  (16x16 * 16x64 -> 4 N sub-tiles, 4 K-steps) ----
#pragma unroll
    for (int t = 0; t < 4; ++t) {
      const int kk = t * 4 + half * 2;
      v2f a;
      a.x = pbuf[w][l16 * 17 + kk];
      a.y = pbuf[w][l16 * 17 + kk + 1];
      const float* vrow0 = V + (bs + key0 + kk) * DIM + hh * DH;
      const float* vrow1 = V + (bs + key0 + kk + 1) * DIM + hh * DH;
#pragma unroll
      for (int nt = 0; nt < 4; ++nt) {
        v2f bf;
        bf.x = vrow0[nt * 16 + l16];
        bf.y = vrow1[nt * 16 + l16];
        acc[nt] = wmma_f32(a, bf, acc[nt]);
      }
    }
  }

  // ---- normalize + merged store (row claimed by matching pattern) ----
#pragma unroll
  for (int r = 0; r < 8; ++r) {
    const int rit = r + half * 8;
    const int row = qbase + rit;
    if ((int)((gqm >> rit) & 1u) == pat) {
      const float inv = 1.f / l_r[r];
#pragma unroll
      for (int nt = 0; nt < 4; ++nt)
        X[(bs + row) * DIM + hh * DH + nt * 16 + l16] = acc[nt][r] * inv;
    }
  }
}

// ---------------------------------------------------------------------------
extern "C" void kernel_launch(void* const* d_in, const int* in_sizes, int n_in,
                              void* d_out, int out_size, void* d_ws,
                              size_t ws_size, hipStream_t stream) {
  const float* xin = (const float*)d_in[0];
  const int* gm    = (const int*)d_in[1];   // bool mask promoted to int32

  float* ws = (float*)d_ws;
  float* proj[6];
  for (int i = 0; i < 6; ++i) proj[i] = ws + (size_t)i * MROWS * DIM;
  float* xattn = ws + (size_t)6 * MROWS * DIM;

  dim3 gg(MROWS / 64, DIM / 32, 1), bb(256);
  // d_in[2+2i]=w, d_in[3+2i]=b for {q_sw,k_sw,v_sw,q_g,k_g,v_g}
  for (int i = 0; i < 6; ++i)
    gemm_512<<<gg, bb, 0, stream>>>(xin, (const float*)d_in[2 + 2 * i],
                                    (const float*)d_in[3 + 2 * i], proj[i]);

  dim3 ga(SEQ / 16 / 4, 2 * NH, 2), ba(128);
  attn_kernel<<<ga, ba, 0, stream>>>(proj[0], proj[1], proj[2], proj[3],
                                     proj[4], proj[5], gm, xattn);

  // y = x @ w_out + b_out   (w_out[h,f,d] is [K=512, N=512] row-major)
  gemm_512<<<gg, bb, 0, stream>>>(xattn, (const float*)d_in[14],
                                  (const float*)d_in[15], (float*)d_out);
}